// VFE_31834297598789
// MI455X (gfx1250) — compile-verified
//
#include <hip/hip_runtime.h>

#define VFE_C 32  // channels per point / per voxel row (32 floats = one 128B cacheline)

// ext_vector float4 so __builtin_nontemporal_load lowers cleanly to
// global_load_b128 with the NT temporal hint on gfx1250.
typedef float vfloat4 __attribute__((ext_vector_type(4)));

// ---------------------------------------------------------------------------
// Scatter-accumulate: 8 lanes cooperate on one point; each lane owns 4 channels.
// - Feature/index reads are NON-TEMPORAL b128/b32 loads: the 512MB stream has
//   zero reuse and must not evict the 25.6MB atomic-accumulation region that
//   we want resident in the 192MB L2.
// - Accumulation uses non-returning global_atomic_add_f32 (L2-resident RMW).
// - Persistent grid-stride form + global_prefetch_b8 one stride ahead hides
//   HBM latency on the streaming loads.
// ---------------------------------------------------------------------------
__global__ void vfe_scatter_kernel(const float* __restrict__ feat,
                                   const int* __restrict__ index,
                                   float* __restrict__ sums,
                                   unsigned int* __restrict__ counts,
                                   int n_points) {
    const int gid    = blockIdx.x * blockDim.x + threadIdx.x;
    const int c4     = (gid & 7) * 4;                  // channel base for this lane
    const int stride = (gridDim.x * blockDim.x) >> 3;  // points per grid pass
    int p = gid >> 3;

    for (; p < n_points; p += stride) {
        // Prefetch the feature line we touch on the next grid-stride pass
        // (speculative; lowers to global_prefetch_b8 on gfx1250).
        const int pn = p + stride;
        if (pn < n_points) {
            __builtin_prefetch(feat + (size_t)pn * VFE_C + c4, 0, 1);
        }

        // Streaming, zero-reuse -> non-temporal load (evict-first in L2).
        const int v = __builtin_nontemporal_load(index + p);  // -1 == masked out
        if (v >= 0) {
            const vfloat4 f = __builtin_nontemporal_load(
                reinterpret_cast<const vfloat4*>(feat + (size_t)p * VFE_C + c4));
            float* dst = sums + (size_t)v * VFE_C + c4;
            atomicAdd(dst + 0, f.x);   // non-returning global_atomic_add_f32
            atomicAdd(dst + 1, f.y);
            atomicAdd(dst + 2, f.z);
            atomicAdd(dst + 3, f.w);
            if (c4 == 0) {
                atomicAdd(counts + v, 1u);  // one count per point
            }
        }
    }
}

// ---------------------------------------------------------------------------
// Finalize: out[v][c] = sums[v][c] / max(count[v], 1).
// Read of sums is L2-hot (RT load); result is written exactly once and never
// re-read on device -> non-temporal b128 store.
// ---------------------------------------------------------------------------
__global__ void vfe_finalize_kernel(float* __restrict__ out,
                                    const unsigned int* __restrict__ counts,
                                    int num_voxels) {
    const int gid   = blockIdx.x * blockDim.x + threadIdx.x;
    const int total = num_voxels * (VFE_C / 4);
    if (gid >= total) return;

    const int v  = gid >> 3;
    const int c4 = (gid & 7) * 4;

    const float inv = 1.0f / fmaxf((float)counts[v], 1.0f);

    vfloat4* p = reinterpret_cast<vfloat4*>(out + (size_t)v * VFE_C + c4);
    vfloat4 s = *p;                 // RT load: sums are L2-hot from the atomics
    s.x *= inv; s.y *= inv; s.z *= inv; s.w *= inv;
    __builtin_nontemporal_store(s, p);   // stream the final result out
}

extern "C" void kernel_launch(void* const* d_in, const int* in_sizes, int n_in,
                              void* d_out, int out_size, void* d_ws, size_t ws_size,
                              hipStream_t stream) {
    const float* feat  = (const float*)d_in[0];   // (N, 32) f32
    const int*   index = (const int*)d_in[1];     // (N,) int
    (void)n_in; (void)ws_size;

    const int n_points   = in_sizes[1];           // N
    const int num_voxels = out_size / VFE_C;      // V (out is V x 32)

    float*        sums   = (float*)d_out;
    unsigned int* counts = (unsigned int*)d_ws;   // V u32 counters in scratch

    // Zero accumulators every call (harness poisons buffers; capture-safe memset).
    hipMemsetAsync(sums, 0, (size_t)out_size * sizeof(float), stream);
    hipMemsetAsync(counts, 0, (size_t)num_voxels * sizeof(unsigned int), stream);

    // Persistent grid-stride scatter: 8192 blocks x 256 threads (8 waves/block),
    // 8 threads per point -> ~16 grid passes over 4M points, prefetch one ahead.
    {
        const int threads = 256;
        int blocks = 8192;
        const long long needed = ((long long)n_points * 8 + threads - 1) / threads;
        if ((long long)blocks > needed) blocks = (int)needed;
        if (blocks < 1) blocks = 1;
        vfe_scatter_kernel<<<blocks, threads, 0, stream>>>(feat, index, sums,
                                                           counts, n_points);
    }

    // Finalize division.
    {
        const int threads = 256;
        const int total   = num_voxels * (VFE_C / 4);
        const int blocks  = (total + threads - 1) / threads;
        vfe_finalize_kernel<<<blocks, threads, 0, stream>>>(sums, counts,
                                                            num_voxels);
    }
}